// Edge_cycle_69827578298774
// MI455X (gfx1250) — compile-verified
//
#include <hip/hip_runtime.h>

// Problem constants (match reference)
#define DD 128
#define NCH 4
#define NEDGE 150000
#define NCYC5 12000
#define SZ5 5
#define NCYC6 10000
#define SZ6 6
#define ROWS5 (NCYC5 * SZ5)          // 60000
#define ROWS6 (NCYC6 * SZ6)          // 60000
#define CTOT (ROWS5 + ROWS6)         // 120000

typedef __attribute__((ext_vector_type(2))) float v2f;
typedef __attribute__((ext_vector_type(8))) float v8f;

__device__ __forceinline__ v8f wmma4(v2f a, v2f b, v8f c) {
    return __builtin_amdgcn_wmma_f32_16x16x4_f32(false, a, false, b,
                                                 (short)0, c, false, false);
}

// ---------------------------------------------------------------------------
// K1: build cat = [cyc | cycle-sum(cyc) | e2c | cycle-sum(e2c)]  (4*D wide)
// ---------------------------------------------------------------------------
__global__ void build_cat_kernel(const float* __restrict__ cyc,
                                 const float* __restrict__ edge,
                                 const int* __restrict__ idx,
                                 float* __restrict__ cat,
                                 int S, int rowBase) {
    int n = blockIdx.x;
    int d = threadIdx.x;                 // 0..127
    float cvals[6], evals[6];
    float csum = 0.f, esum = 0.f;
    for (int s = 0; s < S; ++s) {
        int r = n * S + s;
        float c = cyc[(size_t)r * DD + d];
        const int* ip = idx + (size_t)r * 3;
        float e = edge[(size_t)ip[0] * DD + d]
                + edge[(size_t)ip[1] * DD + d]
                + edge[(size_t)ip[2] * DD + d];
        cvals[s] = c; evals[s] = e;
        csum += c; esum += e;
    }
    for (int s = 0; s < S; ++s) {
        size_t row = (size_t)(rowBase + n * S + s) * (4 * DD);
        cat[row + d]          = cvals[s];
        cat[row + DD + d]     = csum;
        cat[row + 2 * DD + d] = evals[s];
        cat[row + 3 * DD + d] = esum;
    }
}

// ---------------------------------------------------------------------------
// K2: fp32 WMMA GEMM, 48Mx32N register blocking per wave (6 acc tiles).
// C[M,N] = act(A[M,K] @ B[K,N] + bias).  N,K compile-time -> immediate offsets.
// ---------------------------------------------------------------------------
template <int N, int K, int RELU>
__global__ __launch_bounds__(256)
void gemm_wmma48x32(const float* __restrict__ A, const float* __restrict__ B,
                    const float* __restrict__ bias, float* __restrict__ C,
                    int M) {
    constexpr int NTN = N / 32;
    int wave = (int)((blockIdx.x * 256u + threadIdx.x) >> 5);
    int lane = threadIdx.x & 31;
    int total = (M / 48) * NTN;
    if (wave >= total) return;           // wave-uniform
    int tm = (wave / NTN) * 48;
    int tn = (wave % NTN) * 32;
    int hi = lane >> 4, lo = lane & 15;

    const float* a0 = A + (size_t)(tm + lo) * K + 2 * hi;
    const float* a1 = a0 + (size_t)16 * K;
    const float* a2 = a1 + (size_t)16 * K;
    const float* b0 = B + (tn + lo) + 2 * hi * N;
    const float* b1 = b0 + 16;

    v8f acc00 = {}, acc01 = {}, acc10 = {}, acc11 = {}, acc20 = {}, acc21 = {};
    for (int k = 0; k < K; k += 4) {
        v2f fa0 = *(const v2f*)(a0 + k);
        v2f fa1 = *(const v2f*)(a1 + k);
        v2f fa2 = *(const v2f*)(a2 + k);
        v2f fb0, fb1;
        fb0.x = b0[(size_t)k * N];  fb0.y = b0[(size_t)k * N + N];
        fb1.x = b1[(size_t)k * N];  fb1.y = b1[(size_t)k * N + N];
        acc00 = wmma4(fa0, fb0, acc00);
        acc01 = wmma4(fa0, fb1, acc01);
        acc10 = wmma4(fa1, fb0, acc10);
        acc11 = wmma4(fa1, fb1, acc11);
        acc20 = wmma4(fa2, fb0, acc20);
        acc21 = wmma4(fa2, fb1, acc21);
    }

    v8f* accs[3][2] = {{&acc00, &acc01}, {&acc10, &acc11}, {&acc20, &acc21}};
    float bv0 = bias ? bias[tn + lo] : 0.f;
    float bv1 = bias ? bias[tn + 16 + lo] : 0.f;
#pragma unroll
    for (int mi = 0; mi < 3; ++mi) {
        int m0 = tm + 16 * mi + 8 * hi;
#pragma unroll
        for (int ni = 0; ni < 2; ++ni) {
            int bn = tn + 16 * ni + lo;
            float bv = ni ? bv1 : bv0;
            v8f a = *accs[mi][ni];
#pragma unroll
            for (int v = 0; v < 8; ++v) {
                float x = a[v] + bv;
                if (RELU) x = fmaxf(x, 0.f);
                C[(size_t)(m0 + v) * N + bn] = x;
            }
        }
    }
}

// ---------------------------------------------------------------------------
// K3: Autobahn circular conv: out[n,s,:] = relu( sum_j x[n,(s+j)%S,:] @ W[j] )
// Same 48x32 blocking; each lane resolves its rolled source row per shift j.
// ---------------------------------------------------------------------------
template <int S, int DIN, int DOUT>
__global__ __launch_bounds__(256)
void circconv_wmma48x32(const float* __restrict__ X, const float* __restrict__ W,
                        float* __restrict__ OUT, int Ncyc) {
    constexpr int NTN = DOUT / 32;
    int M = Ncyc * S;
    int wave = (int)((blockIdx.x * 256u + threadIdx.x) >> 5);
    int lane = threadIdx.x & 31;
    int total = (M / 48) * NTN;
    if (wave >= total) return;
    int tm = (wave / NTN) * 48;
    int tn = (wave % NTN) * 32;
    int hi = lane >> 4, lo = lane & 15;

    // Per-lane (cycle, slot) for each of the 3 M-subtiles (S is constexpr).
    int ncyc[3], soff[3];
#pragma unroll
    for (int mi = 0; mi < 3; ++mi) {
        int m = tm + 16 * mi + lo;
        ncyc[mi] = m / S;
        soff[mi] = m - ncyc[mi] * S;
    }
    const float* b0 = W + (tn + lo) + 2 * hi * DOUT;
    const float* b1 = b0 + 16;

    v8f acc00 = {}, acc01 = {}, acc10 = {}, acc11 = {}, acc20 = {}, acc21 = {};
#pragma unroll
    for (int j = 0; j < S; ++j) {
        const float* ar[3];
#pragma unroll
        for (int mi = 0; mi < 3; ++mi) {
            int sr = soff[mi] + j; if (sr >= S) sr -= S;   // roll(x,-j) source
            ar[mi] = X + (size_t)(ncyc[mi] * S + sr) * DIN + 2 * hi;
        }
        const float* bj = b0 + (size_t)j * DIN * DOUT;
        const float* bj1 = b1 + (size_t)j * DIN * DOUT;
        for (int k = 0; k < DIN; k += 4) {
            v2f fa0 = *(const v2f*)(ar[0] + k);
            v2f fa1 = *(const v2f*)(ar[1] + k);
            v2f fa2 = *(const v2f*)(ar[2] + k);
            v2f fb0, fb1;
            fb0.x = bj[(size_t)k * DOUT];   fb0.y = bj[(size_t)k * DOUT + DOUT];
            fb1.x = bj1[(size_t)k * DOUT];  fb1.y = bj1[(size_t)k * DOUT + DOUT];
            acc00 = wmma4(fa0, fb0, acc00);
            acc01 = wmma4(fa0, fb1, acc01);
            acc10 = wmma4(fa1, fb0, acc10);
            acc11 = wmma4(fa1, fb1, acc11);
            acc20 = wmma4(fa2, fb0, acc20);
            acc21 = wmma4(fa2, fb1, acc21);
        }
    }

    v8f* accs[3][2] = {{&acc00, &acc01}, {&acc10, &acc11}, {&acc20, &acc21}};
#pragma unroll
    for (int mi = 0; mi < 3; ++mi) {
        int m0 = tm + 16 * mi + 8 * hi;
#pragma unroll
        for (int ni = 0; ni < 2; ++ni) {
            int bn = tn + 16 * ni + lo;
            v8f a = *accs[mi][ni];
#pragma unroll
            for (int v = 0; v < 8; ++v)
                OUT[(size_t)(m0 + v) * DOUT + bn] = fmaxf(a[v], 0.f);
        }
    }
}

// ---------------------------------------------------------------------------
// K4: lin1to1 on cycle outputs -> [x | cycle-sum(x)] (2*D wide)
// ---------------------------------------------------------------------------
__global__ void lin1to1_kernel(const float* __restrict__ X,
                               float* __restrict__ L, int S, int rowBase) {
    int n = blockIdx.x, d = threadIdx.x;
    float vals[6]; float sum = 0.f;
    for (int s = 0; s < S; ++s) {
        vals[s] = X[(size_t)(n * S + s) * DD + d];
        sum += vals[s];
    }
    for (int s = 0; s < S; ++s) {
        size_t row = (size_t)(rowBase + n * S + s) * (2 * DD);
        L[row + d] = vals[s];
        L[row + DD + d] = sum;
    }
}

// ---------------------------------------------------------------------------
// K5: edge concat: ecat = [edge_rep | sum_4 lin[idx][:128] | sum_4 lin[idx][128:]]
// ---------------------------------------------------------------------------
__global__ void edge_cat_kernel(const float* __restrict__ edge,
                                const float* __restrict__ L,
                                const int* __restrict__ idx,
                                float* __restrict__ ecat) {
    int e = blockIdx.x, d = threadIdx.x;
    const int* ip = idx + (size_t)e * 4;
    float s0 = 0.f, s1 = 0.f;
    for (int k = 0; k < 4; ++k) {
        const float* lr = L + (size_t)ip[k] * (2 * DD);
        s0 += lr[d];
        s1 += lr[DD + d];
    }
    size_t row = (size_t)e * (3 * DD);
    ecat[row + d]          = edge[(size_t)e * DD + d];
    ecat[row + DD + d]     = s0;
    ecat[row + 2 * DD + d] = s1;
}

// ---------------------------------------------------------------------------
extern "C" void kernel_launch(void* const* d_in, const int* in_sizes, int n_in,
                              void* d_out, int out_size, void* d_ws, size_t ws_size,
                              hipStream_t stream) {
    const float* edge_rep = (const float*)d_in[0];
    const float* cyc5     = (const float*)d_in[1];
    const float* cyc6     = (const float*)d_in[2];
    const float* W_c1 = (const float*)d_in[3];
    const float* b_c1 = (const float*)d_in[4];
    const float* W_c2 = (const float*)d_in[5];
    const float* b_c2 = (const float*)d_in[6];
    const float* W_e1 = (const float*)d_in[7];
    const float* b_e1 = (const float*)d_in[8];
    const float* W_e2 = (const float*)d_in[9];
    const float* b_e2 = (const float*)d_in[10];
    const float* A1_5 = (const float*)d_in[11];
    const float* A2_5 = (const float*)d_in[12];
    const float* A1_6 = (const float*)d_in[13];
    const float* A2_6 = (const float*)d_in[14];
    const int* e2c5 = (const int*)d_in[15];
    const int* e2c6 = (const int*)d_in[16];
    const int* c2e  = (const int*)d_in[17];

    // Outputs: (edge_out [E,D], out5 [ROWS5,D], out6 [ROWS6,D]) concatenated
    float* edge_out = (float*)d_out;
    float* out5 = edge_out + (size_t)NEDGE * DD;
    float* out6 = out5 + (size_t)ROWS5 * DD;

    // Workspace carve-up (floats)
    float* ws   = (float*)d_ws;
    float* cat  = ws;                              // CTOT * 512
    float* hid  = cat  + (size_t)CTOT * 512;       // CTOT * 256
    float* nw   = hid  + (size_t)CTOT * 256;       // CTOT * 128
    float* mid  = nw   + (size_t)CTOT * 128;       // 60000 * 512 (reused 5 then 6)
    float* lin  = mid  + (size_t)ROWS5 * 512;      // CTOT * 256
    float* ecat = lin  + (size_t)CTOT * 256;       // E * 384
    float* ehid = ecat + (size_t)NEDGE * 384;      // E * 256

    // 1) assemble cycle-MLP input
    build_cat_kernel<<<NCYC5, DD, 0, stream>>>(cyc5, edge_rep, e2c5, cat, SZ5, 0);
    build_cat_kernel<<<NCYC6, DD, 0, stream>>>(cyc6, edge_rep, e2c6, cat, SZ6, ROWS5);

    auto blocks_for = [](int M, int N) {
        int tiles = (M / 48) * (N / 32);
        return (tiles + 7) / 8;                    // 8 waves (256 thr) per block
    };

    // 2) shared cycle MLP: [CTOT,512] -> relu[CTOT,256] -> [CTOT,128]
    gemm_wmma48x32<256, 512, 1><<<blocks_for(CTOT, 256), 256, 0, stream>>>(
        cat, W_c1, b_c1, hid, CTOT);
    gemm_wmma48x32<128, 256, 0><<<blocks_for(CTOT, 128), 256, 0, stream>>>(
        hid, W_c2, b_c2, nw, CTOT);

    // 3) two Autobahn layers per cycle size (relu fused)
    circconv_wmma48x32<SZ5, DD, DD * NCH>
        <<<blocks_for(ROWS5, DD * NCH), 256, 0, stream>>>(nw, A1_5, mid, NCYC5);
    circconv_wmma48x32<SZ5, DD * NCH, DD>
        <<<blocks_for(ROWS5, DD), 256, 0, stream>>>(mid, A2_5, out5, NCYC5);
    circconv_wmma48x32<SZ6, DD, DD * NCH>
        <<<blocks_for(ROWS6, DD * NCH), 256, 0, stream>>>(
            nw + (size_t)ROWS5 * DD, A1_6, mid, NCYC6);
    circconv_wmma48x32<SZ6, DD * NCH, DD>
        <<<blocks_for(ROWS6, DD), 256, 0, stream>>>(mid, A2_6, out6, NCYC6);

    // 4) cycle 1->1 linmaps, scatter-gather back to edges
    lin1to1_kernel<<<NCYC5, DD, 0, stream>>>(out5, lin, SZ5, 0);
    lin1to1_kernel<<<NCYC6, DD, 0, stream>>>(out6, lin, SZ6, ROWS5);
    edge_cat_kernel<<<NEDGE, DD, 0, stream>>>(edge_rep, lin, c2e, ecat);

    // 5) edge MLP: [E,384] -> relu[E,256] -> [E,128] (final output)
    gemm_wmma48x32<256, 384, 1><<<blocks_for(NEDGE, 256), 256, 0, stream>>>(
        ecat, W_e1, b_e1, ehid, NEDGE);
    gemm_wmma48x32<128, 256, 0><<<blocks_for(NEDGE, 128), 256, 0, stream>>>(
        ehid, W_e2, b_e2, edge_out, NEDGE);
}